// SelfCorrelationComputation_26783416057971
// MI455X (gfx1250) — compile-verified
//
#include <hip/hip_runtime.h>
#include <hip/hip_bf16.h>
#include <stdint.h>

// Problem constants (reference: x is (128, 640, 5, 5) f32)
#define B_    128
#define C_    640
#define P_    25          // h*w = 5*5 spatial positions
#define EPSN  1e-12f
#define NC    32          // channels per block in the store kernel
#define XSTR  26          // padded xn row stride (slot 25 == 0 sentinel)

typedef __attribute__((ext_vector_type(2))) float        v2f;
typedef __attribute__((ext_vector_type(4))) float        v4f;
typedef __attribute__((ext_vector_type(8))) float        v8f;
typedef __attribute__((ext_vector_type(4))) unsigned int v4u;
typedef __attribute__((ext_vector_type(4))) int          v4i;
typedef __attribute__((ext_vector_type(8))) int          v8i;

// ---------------------------------------------------------------------------
// Kernel 1: rnorm[b][p] = 1 / max(||relu(x[b,:,p])||_2, eps)
// Sum-of-squares over 640 channels done as the diagonal of Y*Y^T using
// V_WMMA_F32_16X16X4_F32 (full fp32, exact). Two waves per block: wave0 covers
// positions 0..15, wave1 covers 16..24 (zero-padded to 16).
// ---------------------------------------------------------------------------
__global__ __launch_bounds__(64) void rnorm_kernel(const float* __restrict__ x,
                                                   float* __restrict__ rnorm) {
  const int b     = blockIdx.x;
  const int tile  = threadIdx.x >> 5;   // 0 or 1
  const int lane  = threadIdx.x & 31;
  const int half  = lane >> 4;          // lane half selects K group {0,1} vs {2,3}
  const int laneM = lane & 15;
  const int pos   = tile * 16 + laneM;
  const bool pv   = pos < P_;
  const int posc  = pv ? pos : 0;       // clamp to stay in-bounds
  const float* xb = x + (size_t)b * (C_ * P_);

#if __has_builtin(__builtin_amdgcn_wmma_f32_16x16x4_f32)
  v8f acc = {};
  for (int k0 = 0; k0 < C_; k0 += 4) {
    const int ch = k0 + 2 * half;       // K = vgpr + 2*lane_half
    float y0 = xb[(size_t)ch * P_ + posc];
    float y1 = xb[(size_t)(ch + 1) * P_ + posc];
    y0 = pv ? fmaxf(y0, 0.f) : 0.f;     // relu; zero for padded rows
    y1 = pv ? fmaxf(y1, 0.f) : 0.f;
    v2f a; a.x = y0; a.y = y1;
    // Same registers for A and B: D[M,N] += sum_K A[M,K]*A[N,K]  (Gram tile)
    acc = __builtin_amdgcn_wmma_f32_16x16x4_f32(false, a, false, a, (short)0,
                                                acc, false, false);
  }
  // Diagonal: C/D element (M = r + 8*half, N = laneM) -> diag when r+8*half==laneM
  const int r = laneM - 8 * half;
  if (pv && r >= 0 && r < 8) {
    float n2 = 0.f;
#pragma unroll
    for (int rr = 0; rr < 8; ++rr)
      if (rr == r) n2 = acc[rr];
    rnorm[b * P_ + pos] = 1.f / fmaxf(sqrtf(n2), EPSN);
  }
#else
  // Fallback: plain VALU reduction (still correct)
  if (threadIdx.x < P_) {
    const int p = threadIdx.x;
    float s = 0.f;
    for (int c = 0; c < C_; ++c) {
      float v = fmaxf(xb[(size_t)c * P_ + p], 0.f);
      s += v * v;
    }
    rnorm[b * P_ + p] = 1.f / fmaxf(sqrtf(s), EPSN);
  }
#endif
}

// ---------------------------------------------------------------------------
// Kernel 2: out[b,c,j,ki,i,kj] = xn[b,c,ki,kj] * xn_pad[b,c,i+ki-2,j+kj-2]
// One block per (b, 32-channel slab). Input slab (3200 B) pulled into LDS by
// the Tensor Data Mover; 625-entry gather tables make the inner loop
// 2 LDS loads + 1 mul per element; output written with 128-bit NT stores
// (store-bandwidth bound: 204.8 MB total).
// ---------------------------------------------------------------------------
__global__ __launch_bounds__(256) void outer_kernel(const float* __restrict__ x,
                                                    const float* __restrict__ rnorm,
                                                    float* __restrict__ out) {
  __shared__ float         xraw[NC * P_];       // raw input slab (TDM dest)
  __shared__ float         xn[NC * XSTR];       // normalized, padded rows
  __shared__ float         rn[P_];              // 1/norm per position
  __shared__ unsigned char idxA[625];           // ki*5+kj
  __shared__ unsigned char idxB[625];           // gather index or 25 (zero pad)

  const int tid    = threadIdx.x;
  const int b      = blockIdx.x / (C_ / NC);
  const int cchunk = blockIdx.x % (C_ / NC);
  const int c0     = cchunk * NC;
  const float* gsrc = x + (size_t)b * (C_ * P_) + (size_t)c0 * P_;

#if __has_builtin(__builtin_amdgcn_tensor_load_to_lds)
  if (tid < 32) {
    // D# group 0: count=1, lds_addr, 57-bit global addr, type=2 ("image")
    const uint32_t lds_addr = (uint32_t)(uintptr_t)(&xraw[0]);
    const uint64_t ga       = (uint64_t)(uintptr_t)gsrc;
    v4u g0;
    g0[0] = 1u;
    g0[1] = lds_addr;
    g0[2] = (uint32_t)ga;
    g0[3] = (uint32_t)((ga >> 32) & 0x1FFFFFFu) | (2u << 30);
    // D# group 1: data_size=4B (code 2); 1-D tile of 800 elements
    v8i g1;
    g1[0] = 0x20000;                 // wg_mask=0, data_size=2, no flags
    g1[1] = (int)(800u << 16);       // tensor_dim0 = 800 (bits 79:48 low half)
    g1[2] = (int)(1u << 16);         // tensor_dim0 hi=0 | tensor_dim1 = 1
    g1[3] = (int)(800u << 16);       // tensor_dim1 hi=0 | tile_dim0 = 800
    g1[4] = 1;                       // tile_dim1 = 1, tile_dim2 = 0
    g1[5] = 800;                     // tensor_dim0_stride low32
    g1[6] = (int)(800u << 16);       // dim0_stride hi=0 | dim1_stride lo16
    g1[7] = 0;                       // dim1_stride hi32
    v4i gz  = {};
    v8i gz8 = {};
    // 6-arg variant (clang-23 / therock-10.0 headers):
    // (g0, g1, g2, g3, extra_group, cpol)
    __builtin_amdgcn_tensor_load_to_lds(g0, g1, gz, gz, gz8, 0);
  }
#else
  for (int l = tid; l < NC * P_; l += 256) xraw[l] = gsrc[l];
#endif

  // Independent setup while TDM is in flight
  if (tid < P_) rn[tid] = rnorm[b * P_ + tid];
  for (int e = tid; e < 625; e += 256) {
    const int kj = e % 5;
    const int i  = (e / 5) % 5;
    const int ki = (e / 25) % 5;
    const int j  = e / 125;
    idxA[e] = (unsigned char)(ki * 5 + kj);
    const int r = i + ki - 2, s = j + kj - 2;
    idxB[e] = (unsigned char)((r >= 0 && r < 5 && s >= 0 && s < 5) ? (r * 5 + s)
                                                                   : 25);
  }
  if (tid < NC) xn[tid * XSTR + 25] = 0.f;   // zero-pad sentinel slot

#if __has_builtin(__builtin_amdgcn_tensor_load_to_lds)
  if (tid < 32) __builtin_amdgcn_s_wait_tensorcnt(0);
#endif
  __syncthreads();

  // xn = relu(x) * rnorm, laid out with padded rows
  for (int l = tid; l < NC * P_; l += 256) {
    const int c = l / P_, p = l - c * P_;
    xn[c * XSTR + p] = fmaxf(xraw[l], 0.f) * rn[p];
  }
  __syncthreads();

  // 20000 contiguous output floats per block, base 128-B aligned -> v4f NT stores
  float* obase = out + (size_t)(b * C_ + c0) * 625;
  const int NQ = NC * 625 / 4;   // 5000 float4s
  for (int q = tid; q < NQ; q += 256) {
    const int l0 = q * 4;
    int c = l0 / 625;
    int e = l0 - c * 625;
    v4f v;
#pragma unroll
    for (int m = 0; m < 4; ++m) {
      const float* row = &xn[c * XSTR];
      v[m] = row[idxA[e]] * row[idxB[e]];
      if (++e == 625) { e = 0; ++c; }
    }
    __builtin_nontemporal_store(v, (v4f*)(obase + l0));
  }
}

// ---------------------------------------------------------------------------
extern "C" void kernel_launch(void* const* d_in, const int* in_sizes, int n_in,
                              void* d_out, int out_size, void* d_ws,
                              size_t ws_size, hipStream_t stream) {
  (void)in_sizes; (void)n_in; (void)out_size; (void)ws_size;
  const float* x   = (const float*)d_in[0];
  float*       out = (float*)d_out;
  float*       rnm = (float*)d_ws;                 // 128*25 floats = 12.8 KB

  rnorm_kernel<<<B_, 64, 0, stream>>>(x, rnm);
  outer_kernel<<<B_ * (C_ / NC), 256, 0, stream>>>(x, rnm, out);
}